// GCN_34445637714074
// MI455X (gfx1250) — compile-verified
//
#include <hip/hip_runtime.h>
#include <hip/hip_bf16.h>

// ---------------------------------------------------------------------------
// 2-layer GCN for MI455X (gfx1250).
//  - GEMMs via V_WMMA_F32_16X16X4_F32 (exact fp32; GEMM is not the bottleneck)
//  - scatter-aggregation via coalesced float4 gathers + global_atomic_add_f32
//  - self-loop term + bias fused into GEMM epilogue (saves a 51MB pass)
//  - ReLU fused into GEMM2 A-operand loads
// ---------------------------------------------------------------------------

typedef __attribute__((ext_vector_type(2))) float v2f;
typedef __attribute__((ext_vector_type(8))) float v8f;

#define N_NODES 100000
#define N_EDGES 1600000
#define IN_C    128
#define HID_C   128
#define OUT_C   64

// --------------------------- degree / norm kernels -------------------------

__global__ void deg_init_kernel(float* __restrict__ deg, int n) {
    int i = blockIdx.x * blockDim.x + threadIdx.x;
    if (i < n) deg[i] = 1.0f;               // self-loop contributes 1
}

__global__ void deg_count_kernel(const int* __restrict__ dst,
                                 float* __restrict__ deg, int ne) {
    int e = blockIdx.x * blockDim.x + threadIdx.x;
    if (e < ne) atomicAdd(&deg[dst[e]], 1.0f);
}

__global__ void deg_rsqrt_kernel(float* __restrict__ deg, int n) {
    int i = blockIdx.x * blockDim.x + threadIdx.x;
    if (i < n) deg[i] = rsqrtf(deg[i]);     // deg >= 1 always (self-loop)
}

// ------------------------------- WMMA GEMM ---------------------------------
// One wave computes a 16x16 tile of H = A(MxK) * W(KxN), K = 128.
// Epilogue: H stored raw (needed for edge gather), and AGG initialized with
// the self-loop message + bias:  AGG[m][n] = bias[n] + H[m][n] * dis[m]^2.
// RELU applies relu() to the A operand (layer-2 reads relu(agg1)).

template<int NCOLS, bool RELU>
__global__ __launch_bounds__(256)
void gemm_wmma_kernel(const float* __restrict__ A,     // [N_NODES x 128]
                      const float* __restrict__ W,     // [128 x NCOLS]
                      const float* __restrict__ bias,  // [NCOLS]
                      const float* __restrict__ dis,   // [N_NODES]
                      float* __restrict__ H,           // [N_NODES x NCOLS] raw
                      float* __restrict__ AGG)         // [N_NODES x NCOLS] init
{
    constexpr int NT = NCOLS / 16;                     // N tiles
    const int waveId = blockIdx.x * (blockDim.x >> 5) + (threadIdx.x >> 5);
    const int mTile  = waveId / NT;
    const int nTile  = waveId % NT;
    const int lane   = threadIdx.x & 31;
    const int hv     = lane >> 4;                      // half-wave: 0 or 1
    const int l15    = lane & 15;
    const int row    = mTile * 16 + l15;               // A row (M) for this lane
    const int col    = nTile * 16 + l15;               // B/D column (N)

    const float* __restrict__ arow = A + (size_t)row * 128;

    v8f acc = {};
#pragma unroll
    for (int k = 0; k < 128; k += 4) {
        const int ka = k + (hv << 1);                  // lanes 16-31 take K+2,K+3
        float ax = arow[ka];
        float ay = arow[ka + 1];
        if (RELU) { ax = fmaxf(ax, 0.0f); ay = fmaxf(ay, 0.0f); }
        v2f a = { ax, ay };
        v2f b = { W[(size_t)ka * NCOLS + col],
                  W[(size_t)(ka + 1) * NCOLS + col] };
        acc = __builtin_amdgcn_wmma_f32_16x16x4_f32(
            /*neg_a=*/false, a, /*neg_b=*/false, b,
            /*c_mod=*/(short)0, acc, /*reuse_a=*/false, /*reuse_b=*/false);
    }

    // D layout: VGPR v holds (M = mTile*16 + v + 8*hv, N = col)
#pragma unroll
    for (int v = 0; v < 8; ++v) {
        const int m   = mTile * 16 + v + (hv << 3);
        const float d = acc[v];
        const size_t idx = (size_t)m * NCOLS + col;
        H[idx] = d;
        const float s = dis[m];
        AGG[idx] = fmaf(d, s * s, bias[col]);
    }
}

// ----------------------------- edge scatter --------------------------------
// C/4 lanes per edge; each lane gathers a float4 of the source row (coalesced
// global_load_b128) and issues 4 global_atomic_add_f32 to the destination row.

template<int C>
__global__ __launch_bounds__(256)
void scatter_add_kernel(const float* __restrict__ H,
                        const int*   __restrict__ srcs,
                        const int*   __restrict__ dsts,
                        const float* __restrict__ dis,
                        float* __restrict__ AGG, int ne)
{
    constexpr int LPE = C / 4;                          // lanes per edge
    const long long t = (long long)blockIdx.x * blockDim.x + threadIdx.x;
    const int e = (int)(t / LPE);
    const int l = (int)(t % LPE);
    if (e >= ne) return;

    const int s = srcs[e];
    const int d = dsts[e];
    const float norm = dis[s] * dis[d];

    const float4 h4 = ((const float4*)(H + (size_t)s * C))[l];
    float* __restrict__ outp = AGG + (size_t)d * C + (size_t)l * 4;
    atomicAdd(outp + 0, h4.x * norm);
    atomicAdd(outp + 1, h4.y * norm);
    atomicAdd(outp + 2, h4.z * norm);
    atomicAdd(outp + 3, h4.w * norm);
}

// ------------------------------- launcher ----------------------------------

static inline size_t align256(size_t x) { return (x + 255) & ~(size_t)255; }

extern "C" void kernel_launch(void* const* d_in, const int* in_sizes, int n_in,
                              void* d_out, int out_size, void* d_ws, size_t ws_size,
                              hipStream_t stream) {
    (void)in_sizes; (void)n_in; (void)out_size; (void)ws_size;

    const float* x    = (const float*)d_in[0];           // [N_NODES,128]
    const int*   ei   = (const int*)  d_in[1];           // [2, N_EDGES]
    const float* W1   = (const float*)d_in[2];           // [128,128]
    const float* b1   = (const float*)d_in[3];           // [128]
    const float* W2   = (const float*)d_in[4];           // [128,64]
    const float* b2   = (const float*)d_in[5];           // [64]
    float*       out  = (float*)d_out;                   // [N_NODES,64]

    const int* src = ei;                                  // edge_index[0]
    const int* dst = ei + N_EDGES;                        // edge_index[1]

    // workspace layout
    char* ws = (char*)d_ws;
    size_t off = 0;
    float* dis  = (float*)(ws + off); off = align256(off + sizeof(float) * N_NODES);
    float* h1   = (float*)(ws + off); off = align256(off + sizeof(float) * (size_t)N_NODES * HID_C);
    float* agg1 = (float*)(ws + off); off = align256(off + sizeof(float) * (size_t)N_NODES * HID_C);
    float* h2   = (float*)(ws + off); off = align256(off + sizeof(float) * (size_t)N_NODES * OUT_C);

    // 1-3) degrees -> dis = rsqrt(deg)
    deg_init_kernel<<<(N_NODES + 255) / 256, 256, 0, stream>>>(dis, N_NODES);
    deg_count_kernel<<<(N_EDGES + 255) / 256, 256, 0, stream>>>(dst, dis, N_EDGES);
    deg_rsqrt_kernel<<<(N_NODES + 255) / 256, 256, 0, stream>>>(dis, N_NODES);

    // 4) h1 = x @ W1 ; agg1 = b1 + h1*dis^2   (self-loop + bias fused)
    {
        const int waves  = (N_NODES / 16) * (HID_C / 16);  // 6250 * 8
        const int blocks = waves / 8;                      // 8 waves / block
        gemm_wmma_kernel<HID_C, false><<<blocks, 256, 0, stream>>>(
            x, W1, b1, dis, h1, agg1);
    }

    // 5) agg1[dst] += h1[src] * dis[src]*dis[dst]   (C=128: 1 wave/edge)
    {
        const long long thr = (long long)N_EDGES * (HID_C / 4);
        scatter_add_kernel<HID_C><<<(int)((thr + 255) / 256), 256, 0, stream>>>(
            h1, src, dst, dis, agg1, N_EDGES);
    }

    // 6) h2 = relu(agg1) @ W2 ; out = b2 + h2*dis^2
    {
        const int waves  = (N_NODES / 16) * (OUT_C / 16);  // 6250 * 4
        const int blocks = waves / 8;
        gemm_wmma_kernel<OUT_C, true><<<blocks, 256, 0, stream>>>(
            agg1, W2, b2, dis, h2, out);
    }

    // 7) out[dst] += h2[src] * dis[src]*dis[dst]   (C=64: half-wave/edge)
    {
        const long long thr = (long long)N_EDGES * (OUT_C / 4);
        scatter_add_kernel<OUT_C><<<(int)((thr + 255) / 256), 256, 0, stream>>>(
            h2, src, dst, dis, out, N_EDGES);
    }
}